// GCNLayer_83133386981887
// MI455X (gfx1250) — compile-verified
//
#include <hip/hip_runtime.h>
#include <cstdint>
#include <cstddef>

typedef unsigned int u32;
typedef u32 u32x4 __attribute__((ext_vector_type(4)));
typedef int  i32x4 __attribute__((ext_vector_type(4)));
typedef int  i32x8 __attribute__((ext_vector_type(8)));

// LDS (addrspace 3) pointer types so the staged-edge reads lower to ds_load_b32,
// not flat_load. volatile: the only writer is the TDM, which the compiler can't see.
typedef __attribute__((address_space(3))) const volatile int   lds_cvint;
typedef __attribute__((address_space(3))) const volatile float lds_cvfloat;

#define WPB   4            // waves per block (wave32)
#define TPB   (WPB * 32)
#define CHUNK 512          // edges staged per TDM op per wave

// ---------------------------------------------------------------------------
// TDM: DMA a 1-D run of CHUNK 32-bit elements from global memory into LDS.
// Descriptor layout per CDNA5 ISA ch.8 (D# group0 = 4 dwords, group1 = 8 dwords).
// tensor_dim0 = remaining elements => tail reads are OOB and return zero.
// ---------------------------------------------------------------------------
__device__ __forceinline__ void tdm_load_1d_b32(u32 lds_byte_off, const void* gsrc,
                                                u32 remaining_elems, u32 tile_elems) {
  uint64_t ga = (uint64_t)(uintptr_t)gsrc;
  u32x4 g0;
  g0.x = 1u;                                           // count=1 (valid user D#)
  g0.y = lds_byte_off;                                 // lds_addr (bytes)
  g0.z = (u32)ga;                                      // global_addr[31:0]
  g0.w = (u32)((ga >> 32) & 0x01FFFFFFu) | (2u << 30); // global_addr[56:32] | type=2

  i32x8 g1;
  g1[0] = (int)(2u << 16);                                  // data_size=4B, wg_mask=0
  g1[1] = (int)((remaining_elems & 0xFFFFu) << 16);         // tensor_dim0[15:0]
  g1[2] = (int)((remaining_elems >> 16) | (1u << 16));      // tensor_dim0[31:16] | tensor_dim1=1
  g1[3] = (int)(tile_elems << 16);                          // tile_dim0
  g1[4] = 1;                                                // tile_dim1=1, tile_dim2=0
  g1[5] = (int)tile_elems;                                  // tensor_dim0_stride[31:0]
  g1[6] = 0;                                                // stride0[47:32] | stride1[15:0]
  g1[7] = 0;                                                // stride1[47:16]

  // Groups 2/3: inner dims never stepped (tile_dim2=0); keep dims non-zero.
  i32x4 g2; g2[0] = 1; g2[1] = 1; g2[2] = 0; g2[3] = 0;     // tensor_dim2=1, tensor_dim3=1
  i32x4 g3; g3[0] = 0; g3[1] = 1 << 16; g3[2] = 0; g3[3] = 0; // tensor_dim4=1

#if __has_builtin(__builtin_amdgcn_tensor_load_to_lds)
# if defined(__clang_major__) && (__clang_major__ >= 23)
  i32x8 g4 = {0, 0, 0, 0, 0, 0, 0, 0};
  __builtin_amdgcn_tensor_load_to_lds(g0, g1, g2, g3, g4, 0);
# else
  __builtin_amdgcn_tensor_load_to_lds(g0, g1, g2, g3, 0);
# endif
#else
  asm volatile("tensor_load_to_lds %0, %1, %2, %3"
               :: "s"(g0), "s"(g1), "s"(g2), "s"(g3) : "memory");
#endif
}

__device__ __forceinline__ void wait_tensorcnt_le(int n) {
#if __has_builtin(__builtin_amdgcn_s_wait_tensorcnt)
  if (n == 0) __builtin_amdgcn_s_wait_tensorcnt(0);
  else        __builtin_amdgcn_s_wait_tensorcnt(3);
#else
  if (n == 0) asm volatile("s_wait_tensorcnt 0x0" ::: "memory");
  else        asm volatile("s_wait_tensorcnt 0x3" ::: "memory");
#endif
  asm volatile("" ::: "memory");  // compiler barrier: don't hoist LDS reads
}

// ---------------------------------------------------------------------------
// Zero the output (atomics accumulate into it).
// ---------------------------------------------------------------------------
__global__ void gcn_zero_kernel(float4* __restrict__ out, u32 n4) {
  u32 i = blockIdx.x * blockDim.x + threadIdx.x;
  u32 stride = gridDim.x * blockDim.x;
  float4 z = make_float4(0.f, 0.f, 0.f, 0.f);
  for (; i < n4; i += stride) out[i] = z;
}

// ---------------------------------------------------------------------------
// Scatter-add: each wave owns CHUNK-edge chunks (strided), double-buffers the
// edge stream into LDS with the TDM, then for each edge all 32 lanes gather one
// 512B embeds row (float4/lane, L2-resident) and atomically add into out[row].
// ---------------------------------------------------------------------------
__global__ __launch_bounds__(TPB) void gcn_scatter_kernel(
    const int* __restrict__ rows, const int* __restrict__ cols,
    const float* __restrict__ vals, const float* __restrict__ embeds,
    float* __restrict__ out, u32 E, u32 nchunks, u32 waveStride)
{
  __shared__ int   s_rows[WPB * 2 * CHUNK];
  __shared__ int   s_cols[WPB * 2 * CHUNK];
  __shared__ float s_vals[WPB * 2 * CHUNK];

  const u32 lane   = threadIdx.x & 31u;
  const u32 wlocal = threadIdx.x >> 5;
  const u32 wave   = (u32)__builtin_amdgcn_readfirstlane((int)(blockIdx.x * WPB + wlocal));

  const u32 lds_rows_base = (u32)(uintptr_t)(void*)s_rows;
  const u32 lds_cols_base = (u32)(uintptr_t)(void*)s_cols;
  const u32 lds_vals_base = (u32)(uintptr_t)(void*)s_vals;

  const float4* __restrict__ embv = (const float4*)embeds;

  auto issue = [&](u32 ck, u32 buf) {
    const u32 base = ck * CHUNK;
    const u32 rem  = E - base;                      // bound for OOB-zero tail
    const u32 off  = ((wlocal * 2u + buf) * CHUNK) * 4u;
    tdm_load_1d_b32(lds_rows_base + off, rows + base, rem, CHUNK);
    tdm_load_1d_b32(lds_cols_base + off, cols + base, rem, CHUNK);
    tdm_load_1d_b32(lds_vals_base + off, vals + base, rem, CHUNK);
  };

  u32 c = wave;
  if (c >= nchunks) return;

  issue(c, 0);                                      // prime buffer 0
  u32 it = 0;
  for (; c < nchunks; c += waveStride, ++it) {
    const u32  cn   = c + waveStride;
    const bool more = (cn < nchunks);
    if (more) {
      asm volatile("s_wait_dscnt 0x0" ::: "memory"); // LDS reads done before TDM overwrite
      issue(cn, (it + 1u) & 1u);
    }
    wait_tensorcnt_le(more ? 3 : 0);                // chunk c's 3 TDM ops complete (in-order)

    const u32 buf  = it & 1u;
    const u32 sidx = (wlocal * 2u + buf) * CHUNK;
    lds_cvint*   sr = (lds_cvint*)  (&s_rows[sidx]);  // ds_load_b32 path
    lds_cvint*   sc = (lds_cvint*)  (&s_cols[sidx]);
    lds_cvfloat* sv = (lds_cvfloat*)(&s_vals[sidx]);

    const u32 base  = c * CHUNK;
    u32 count = E - base;
    if (count > CHUNK) count = CHUNK;

    for (u32 e = 0; e < count; ++e) {
      const int   r   = sr[e];
      const int   col = sc[e];
      const float w   = sv[e];
      float4 v = embv[(size_t)col * 32u + lane];    // global_load_b128, coalesced 512B/row
      float* dst = out + (size_t)r * 128u + lane * 4u;
      unsafeAtomicAdd(dst + 0, v.x * w);            // global_atomic_add_f32 (L2-resident)
      unsafeAtomicAdd(dst + 1, v.y * w);
      unsafeAtomicAdd(dst + 2, v.z * w);
      unsafeAtomicAdd(dst + 3, v.w * w);
    }
  }
}

// ---------------------------------------------------------------------------
// d_in: [0] adj_indices int32 [2,E] (rows then cols), [1] adj_values f32 [E],
//       [2] embeds f32 [N,128], [3] batch_size (unused).
// d_out: f32 [N,128].
// ---------------------------------------------------------------------------
extern "C" void kernel_launch(void* const* d_in, const int* in_sizes, int n_in,
                              void* d_out, int out_size, void* d_ws, size_t ws_size,
                              hipStream_t stream) {
  (void)n_in; (void)d_ws; (void)ws_size;
  const int*   adj    = (const int*)d_in[0];
  const float* vals   = (const float*)d_in[1];
  const float* embeds = (const float*)d_in[2];
  const u32    E      = (u32)in_sizes[1];
  const int*   rows   = adj;
  const int*   cols   = adj + E;
  float*       out    = (float*)d_out;

  // 1) zero the accumulator
  const u32 n4 = (u32)out_size / 4u;
  gcn_zero_kernel<<<2048, 256, 0, stream>>>((float4*)out, n4);

  // 2) TDM-staged gather/scatter-add
  const u32 nchunks    = (E + CHUNK - 1u) / CHUNK;
  u32 blocks           = (nchunks + 2u * WPB - 1u) / (2u * WPB); // ~2 chunks/wave
  if (blocks == 0) blocks = 1;
  const u32 waveStride = blocks * WPB;
  gcn_scatter_kernel<<<blocks, TPB, 0, stream>>>(rows, cols, vals, embeds, out,
                                                 E, nchunks, waveStride);
}